// CRFLayer_27092653703713
// MI455X (gfx1250) — compile-verified
//
#include <hip/hip_runtime.h>
#include <hip/hip_bf16.h>
#include <stdint.h>

#define NCLS 21
static constexpr int BLOCK = 256;

static inline int nblocks(long n) { return (int)((n + BLOCK - 1) / BLOCK); }

// ---------------------------------------------------------------- blur
// dst[m,c] = src[m,c] + 0.5*(src[n1[m],c] + src[n2[m],c]), m < M.
// Row M of src is the zero sentinel for missing neighbors (index == M).
// The contiguous center stream is staged through LDS with the CDNA5 async
// global->LDS path (ASYNCcnt), published with s_wait_asynccnt + barrier,
// and consumed with an explicit ds_load_b32 (DScnt). The irregular
// neighbor gathers overlap the async copy.
template<int C>
__device__ __forceinline__ void blur_body(const float* __restrict__ src, float* __restrict__ dst,
                                          const int* __restrict__ n1, const int* __restrict__ n2,
                                          int M) {
    __shared__ __align__(16) float s_center[BLOCK];
    const int total = M * C;
    const int base  = blockIdx.x * BLOCK;
    const int tid   = threadIdx.x;
    const int idx   = base + tid;
    const bool full = (base + BLOCK) <= total;   // block-uniform

    if (full && tid < BLOCK / 4) {
        uint32_t lds   = (uint32_t)(uintptr_t)(void*)&s_center[tid * 4];
        uint64_t gaddr = (uint64_t)(uintptr_t)(const void*)(src + base + tid * 4);
        asm volatile("global_load_async_to_lds_b128 %0, %1, off"
                     :: "v"(lds), "v"(gaddr) : "memory");
    }

    float a = 0.0f, b = 0.0f;
    if (idx < total) {
        int m = idx / C;
        int c = idx - m * C;
        a = src[n1[m] * C + c];
        b = src[n2[m] * C + c];
    }

    asm volatile("s_wait_asynccnt 0" ::: "memory");   // issuing waves: copy done
    __syncthreads();                                  // publish LDS to all waves

    if (idx < total) {
        float center;
        if (full) {
            uint32_t lds = (uint32_t)(uintptr_t)(void*)&s_center[tid];
            asm volatile("ds_load_b32 %0, %1\n\t"
                         "s_wait_dscnt 0"
                         : "=v"(center) : "v"(lds) : "memory");
        } else {
            center = src[idx];
        }
        dst[idx] = center + 0.5f * (a + b);
    }
}

__global__ void blur21_kernel(const float* __restrict__ src, float* __restrict__ dst,
                              const int* __restrict__ n1, const int* __restrict__ n2, int M) {
    blur_body<NCLS>(src, dst, n1, n2, M);
}

__global__ void blur1_kernel(const float* __restrict__ src, float* __restrict__ dst,
                             const int* __restrict__ n1, const int* __restrict__ n2, int M) {
    blur_body<1>(src, dst, n1, n2, M);
}

// ---------------------------------------------------------------- zero fill
__global__ void zero_kernel(float* __restrict__ p, int n) {
    int i = blockIdx.x * blockDim.x + threadIdx.x;
    if (i < n) p[i] = 0.0f;
}

// ---------------------------------------------------------------- splat
template<int L>
__global__ void splat_ones_kernel(const int* __restrict__ o, const float* __restrict__ w,
                                  float* __restrict__ vals, int npix) {
    int t = blockIdx.x * blockDim.x + threadIdx.x;
    if (t >= npix * L) return;
    atomicAdd(&vals[o[t]], w[t]);
}

template<int C, int L>
__global__ void splat_kernel(const float* __restrict__ Q, const float* __restrict__ norm,
                             const int* __restrict__ o, const float* __restrict__ w,
                             float* __restrict__ vals, int npix) {
    int t = blockIdx.x * blockDim.x + threadIdx.x;
    if (t >= npix * C) return;
    int n = t / C;
    int c = t - n * C;
    float q = Q[t] * norm[n];
    #pragma unroll
    for (int l = 0; l < L; ++l) {
        int   m  = o[n * L + l];
        float ww = w[n * L + l];
        atomicAdd(&vals[m * C + c], ww * q);
    }
}

// ---------------------------------------------------------------- slice
template<int L>
__global__ void slice_norm_kernel(const int* __restrict__ o, const float* __restrict__ w,
                                  const float* __restrict__ vals, float* __restrict__ norm,
                                  int npix, float alpha) {
    int n = blockIdx.x * blockDim.x + threadIdx.x;
    if (n >= npix) return;
    float s = 0.0f;
    #pragma unroll
    for (int l = 0; l < L; ++l)
        s += w[n * L + l] * vals[o[n * L + l]];
    norm[n] = 1.0f / (sqrtf(alpha * s) + 1e-20f);
}

template<int C, int L, bool ACCUM>
__global__ void slice_kernel(const int* __restrict__ o, const float* __restrict__ w,
                             const float* __restrict__ vals, const float* __restrict__ norm,
                             float* __restrict__ acc, int npix, float scale) {
    int t = blockIdx.x * blockDim.x + threadIdx.x;
    if (t >= npix * C) return;
    int n = t / C;
    int c = t - n * C;
    float s = 0.0f;
    #pragma unroll
    for (int l = 0; l < L; ++l)
        s += w[n * L + l] * vals[o[n * L + l] * C + c];
    float r = scale * s * norm[n];
    acc[t] = ACCUM ? (acc[t] + r) : r;
}

// ---------------------------------------------------------------- softmax update
// Q = softmax(-unary + acc)   (pairwise = -message; -u - pairwise = -u + message)
template<int C>
__global__ void softmax_kernel(const float* __restrict__ unary, const float* __restrict__ acc,
                               float* __restrict__ Q, int npix, int use_acc) {
    int n = blockIdx.x * blockDim.x + threadIdx.x;
    if (n >= npix) return;
    float v[C];
    float mx = -3.4e38f;
    #pragma unroll
    for (int c = 0; c < C; ++c) {
        float x = -unary[n * C + c];
        if (use_acc) x += acc[n * C + c];
        v[c] = x;
        mx = fmaxf(mx, x);
    }
    float sum = 0.0f;
    #pragma unroll
    for (int c = 0; c < C; ++c) {
        v[c] = expf(v[c] - mx);
        sum += v[c];
    }
    float inv = 1.0f / sum;
    #pragma unroll
    for (int c = 0; c < C; ++c)
        Q[n * C + c] = v[c] * inv;
}

// ---------------------------------------------------------------- host helpers
template<int L>
static void run_norm(const int* o, const float* w, const int* n1, const int* n2,
                     int M, int npix, float alpha,
                     float* vA, float* vB, float* norm, hipStream_t stream) {
    zero_kernel<<<nblocks(M + 1), BLOCK, 0, stream>>>(vA, M + 1);
    zero_kernel<<<nblocks(M + 1), BLOCK, 0, stream>>>(vB, M + 1);
    splat_ones_kernel<L><<<nblocks((long)npix * L), BLOCK, 0, stream>>>(o, w, vA, npix);
    float* bufs[2] = { vA, vB };
    int cur = 0;
    for (int j = 0; j < L; ++j) {
        blur1_kernel<<<nblocks(M), BLOCK, 0, stream>>>(
            bufs[cur], bufs[cur ^ 1], n1 + (long)j * M, n2 + (long)j * M, M);
        cur ^= 1;
    }
    slice_norm_kernel<L><<<nblocks(npix), BLOCK, 0, stream>>>(o, w, bufs[cur], norm, npix, alpha);
}

template<int C, int L, bool ACCUM>
static void run_filter(const float* Q, const float* norm,
                       const int* o, const float* w, const int* n1, const int* n2,
                       int M, int npix, float scale,
                       float* vA, float* vB, float* acc, hipStream_t stream) {
    // vA fully zero (splat accumulator + sentinel); vB only needs sentinel row zero
    zero_kernel<<<nblocks((long)(M + 1) * C), BLOCK, 0, stream>>>(vA, (M + 1) * C);
    zero_kernel<<<1, BLOCK, 0, stream>>>(vB + (long)M * C, C);
    splat_kernel<C, L><<<nblocks((long)npix * C), BLOCK, 0, stream>>>(Q, norm, o, w, vA, npix);
    float* bufs[2] = { vA, vB };
    int cur = 0;
    for (int j = 0; j < L; ++j) {
        blur21_kernel<<<nblocks((long)M * C), BLOCK, 0, stream>>>(
            bufs[cur], bufs[cur ^ 1], n1 + (long)j * M, n2 + (long)j * M, M);
        cur ^= 1;
    }
    slice_kernel<C, L, ACCUM><<<nblocks((long)npix * C), BLOCK, 0, stream>>>(
        o, w, bufs[cur], norm, acc, npix, scale);
}

// ---------------------------------------------------------------- entry point
extern "C" void kernel_launch(void* const* d_in, const int* in_sizes, int n_in,
                              void* d_out, int out_size, void* d_ws, size_t ws_size,
                              hipStream_t stream) {
    const float* unary = (const float*)d_in[0];
    const float* bw    = (const float*)d_in[1];
    const float* sw    = (const float*)d_in[2];
    const int*   bo    = (const int*)  d_in[3];
    const int*   bn1   = (const int*)  d_in[4];
    const int*   bn2   = (const int*)  d_in[5];
    const int*   so    = (const int*)  d_in[6];
    const int*   sn1   = (const int*)  d_in[7];
    const int*   sn2   = (const int*)  d_in[8];

    const int npix = in_sizes[0] / NCLS;          // 512*512
    const int Mb   = in_sizes[4] / 6;             // bilateral lattice vertices
    const int Ms   = in_sizes[7] / 3;             // spatial lattice vertices
    const int Mmax = (Mb > Ms) ? Mb : Ms;

    // carve workspace (256B aligned chunks so async b128 staging stays 16B aligned)
    size_t off = 0;
    auto carve = [&](long nfloats) -> float* {
        float* p = (float*)((char*)d_ws + off);
        off += (size_t)(((nfloats * 4 + 255) / 256) * 256);
        return p;
    };
    float* bnorm = carve(npix);
    float* snorm = carve(npix);
    float* acc   = carve((long)npix * NCLS);
    float* valsA = carve((long)(Mmax + 1) * NCLS);
    float* valsB = carve((long)(Mmax + 1) * NCLS);
    float* Q     = (float*)d_out;

    const float alpha_b = 32.0f / 33.0f;  // 1/(1+2^-5), d=5
    const float alpha_s = 0.8f;           // 1/(1+2^-2), d=2

    // normalization filters (filter of ones)
    run_norm<6>(bo, bw, bn1, bn2, Mb, npix, alpha_b, valsA, valsB, bnorm, stream);
    run_norm<3>(so, sw, sn1, sn2, Ms, npix, alpha_s, valsA, valsB, snorm, stream);

    // Q0 = softmax(-unary)
    softmax_kernel<NCLS><<<nblocks(npix), BLOCK, 0, stream>>>(unary, nullptr, Q, npix, 0);

    for (int it = 0; it < 5; ++it) {
        run_filter<NCLS, 6, false>(Q, bnorm, bo, bw, bn1, bn2, Mb, npix,
                                   alpha_b * 10.0f, valsA, valsB, acc, stream);
        run_filter<NCLS, 3, true >(Q, snorm, so, sw, sn1, sn2, Ms, npix,
                                   alpha_s * 3.0f, valsA, valsB, acc, stream);
        softmax_kernel<NCLS><<<nblocks(npix), BLOCK, 0, stream>>>(unary, acc, Q, npix, 1);
    }
}